// CRF_19353122636065
// MI455X (gfx1250) — compile-verified
//
#include <hip/hip_runtime.h>

#define NLAB 128
#define TSTEPS 2048
#define NBATCH 128
#define START_LABEL 126
#define STOP_LABEL 127
#define RING 8            // async emission ring slots (power of 2)
#define AHEAD 7           // prefetch depth = RING-1; s_wait_asynccnt immediate

typedef float v2f __attribute__((ext_vector_type(2)));
typedef float v4f __attribute__((ext_vector_type(4)));

__global__ __launch_bounds__(256) void crf_viterbi_kernel(
    const float* __restrict__ emissions,   // [B, T, L] f32
    const float* __restrict__ transitions, // [L, L]    f32
    float* __restrict__ out)               // [2*B]: scores, then labels
{
    __shared__ __attribute__((aligned(16))) float v_sh[NLAB];
    __shared__ __attribute__((aligned(16))) float pm_sh[256];
    __shared__ __attribute__((aligned(16))) float em_sh[RING][NLAB];  // async ring

    const int tid = threadIdx.x;
    const int b   = blockIdx.x;
    const int j   = tid & (NLAB - 1);
    const int h   = tid >> 7;          // 0 or 1: which half of the k-range
    const int kbase = h * 64;

    // ---- hoist this thread's transition column slice into VGPRs ----
    // thread (j,h) only ever needs trans[kbase+k][j], k=0..63. Stored as
    // <2 x float> pairs so the inner-loop adds can lower to v_pk_add_f32.
    v2f ta2[32];
    {
        const float* tp = transitions + (size_t)kbase * NLAB + j;
        #pragma unroll
        for (int k2 = 0; k2 < 32; ++k2) {
            v2f t;
            t.x = tp[(size_t)(2 * k2 + 0) * NLAB];   // 64-load clause, imm offsets
            t.y = tp[(size_t)(2 * k2 + 1) * NLAB];
            ta2[k2] = t;
        }
    }
    // trans[:, STOP] needed once at the end; keep in a register.
    float tstop = 0.0f;
    if (tid < NLAB)
        tstop = transitions[(size_t)tid * NLAB + STOP_LABEL];

    if (tid < NLAB)
        v_sh[tid] = (tid == START_LABEL) ? 0.0f : -10000.0f;

    const size_t   em_base     = ((size_t)b * TSTEPS) * NLAB;
    const unsigned lds_em_base = (unsigned)(unsigned long long)(&em_sh[0][0]);

    // ---- async prefetch emission rows t = 1..AHEAD into ring slots ----
    if (tid < NLAB) {
        #pragma unroll
        for (int tt = 1; tt <= AHEAD; ++tt) {
            unsigned loff = lds_em_base + (unsigned)(((tt & (RING - 1)) * NLAB + tid) * 4);
            const float* gp = emissions + em_base + (size_t)tt * NLAB + tid;
            asm volatile("global_load_async_to_lds_b32 %0, %1, off"
                         :: "v"(loff), "v"(gp) : "memory");
        }
    }
    __syncthreads();

    for (int t = 1; t < TSTEPS; ++t) {
        // Prefetch row min(t+AHEAD, T-1) into slot (t+AHEAD)&(RING-1). Rows
        // beyond the end are clamped junk into slots never read again (keeps
        // the ASYNCcnt pipeline depth invariant so the wait stays constant).
        if (tid < NLAB) {
            int pt = t + AHEAD; if (pt > TSTEPS - 1) pt = TSTEPS - 1;
            unsigned loff = lds_em_base +
                (unsigned)((((t + AHEAD) & (RING - 1)) * NLAB + tid) * 4);
            const float* gp = emissions + em_base + (size_t)pt * NLAB + tid;
            asm volatile("global_load_async_to_lds_b32 %0, %1, off"
                         :: "v"(loff), "v"(gp) : "memory");
        }

        // ---- partial max-plus over k in [kbase, kbase+64), trans in VGPRs ----
        // Adds as <2 x float> fadd -> v_pk_add_f32 (VOP3P); maxes scalarize to
        // independent v_max_num_f32 pairs, eligible for VOPD dual-issue.
        const float NEG = -3.402823466e+38f;
        v2f M0 = {NEG, NEG}, M1 = {NEG, NEG};
        #pragma unroll
        for (int k4 = 0; k4 < 16; ++k4) {
            v4f vv = *reinterpret_cast<const v4f*>(&v_sh[kbase + k4 * 4]); // bcast
            v2f s01 = vv.xy + ta2[2 * k4 + 0];   // v_pk_add_f32
            v2f s23 = vv.zw + ta2[2 * k4 + 1];   // v_pk_add_f32
            M0 = __builtin_elementwise_max(M0, s01);
            M1 = __builtin_elementwise_max(M1, s23);
        }
        v2f Mt = __builtin_elementwise_max(M0, M1);
        pm_sh[tid] = fmaxf(Mt.x, Mt.y);
        __syncthreads();

        if (tid < NLAB) {
            // Oldest of <=RING in-flight async loads is row t; each wave reads
            // only the em bytes it loaded itself, so a per-wave wait suffices.
            asm volatile("s_wait_asynccnt 7" ::: "memory");
            float vn = fmaxf(pm_sh[tid], pm_sh[tid + NLAB]) +
                       em_sh[t & (RING - 1)][tid];
            v_sh[tid] = vn;
        }
        __syncthreads();
    }

    // ---- finalize: add transition into STOP, then max / first-argmax ----
    if (tid < NLAB)
        pm_sh[tid] = v_sh[tid] + tstop;
    __syncthreads();
    if (tid == 0) {
        float best = pm_sh[0];
        int bestj = 0;
        #pragma unroll 1   // once-per-block scan; keep it a compact loop
        for (int jj = 1; jj < NLAB; ++jj) {
            float x = pm_sh[jj];
            if (x > best) { best = x; bestj = jj; }  // strict '>' => first max (argmax)
        }
        out[b] = best;
        out[NBATCH + b] = (float)bestj;
    }
}

extern "C" void kernel_launch(void* const* d_in, const int* in_sizes, int n_in,
                              void* d_out, int out_size, void* d_ws, size_t ws_size,
                              hipStream_t stream) {
    const float* emissions   = (const float*)d_in[0]; // [128, 2048, 128]
    const float* transitions = (const float*)d_in[1]; // [128, 128]
    float* out = (float*)d_out;                       // [256]: scores[128], labels[128]
    (void)in_sizes; (void)n_in; (void)out_size; (void)d_ws; (void)ws_size;
    crf_viterbi_kernel<<<dim3(NBATCH), dim3(256), 0, stream>>>(emissions, transitions, out);
}